// HnetSparse_75067438399650
// MI455X (gfx1250) — compile-verified
//
#include <hip/hip_runtime.h>

// CDNA5 / gfx1250. wave32. WMMA f32 16x16x4 fragments:
//   A 16x4:  VGPR0: lanes0-15 K=0, lanes16-31 K=2 ; VGPR1: K=1 / K=3
//   B 4x16:  mirrors A (lanes0-15 hold K=0,1; lanes16-31 hold K=2,3), N = lane&15
//   C/D 16x16: VGPR r: lanes0-15 (M=r, N=lane), lanes16-31 (M=r+8, N=lane-16)
typedef __attribute__((ext_vector_type(2))) float v2f;
typedef __attribute__((ext_vector_type(4))) float v4f;
typedef __attribute__((ext_vector_type(8))) float v8f;

__device__ __forceinline__ int lower_bound_i32(const int* __restrict__ a, int n, int key) {
  int lo = 0, hi = n;
  while (lo < hi) {
    int mid = (lo + hi) >> 1;
    if (a[mid] < key) lo = mid + 1; else hi = mid;
  }
  return lo;
}

// One wave per 16-row output tile. blockDim = 256 (8 waves).
// h   : input activations, [in_rows][32] row-major
// out : if !tstore: [out_size][32] row-major (optionally ReLU'd)
//       if  tstore: [32][out_size]  (final transposed store through LDS, non-temporal)
__global__ __launch_bounds__(256)
void hnet_spmm_wmma(const float* __restrict__ h,
                    const float* __restrict__ val,
                    const int*   __restrict__ io,
                    const int*   __restrict__ ii,
                    float*       __restrict__ out,
                    int out_size, int nnz, int do_relu, int tstore)
{
  // Per-wave staged tile metadata, padded to 52 entries (max j reach = kcnt+2 <= 50):
  // padding entries carry io = -1 (matches no row -> A element 0) and a valid ii
  // (so B gather addresses stay in range; the garbage row is multiplied by zero).
  __shared__ int   io_s [8][52];
  __shared__ int   ii_s [8][52];
  __shared__ float val_s[8][52];
  __shared__ float xpose[8][16][33];

  const int lane  = threadIdx.x & 31;
  const int wslot = threadIdx.x >> 5;
  const int O     = (blockIdx.x * 8 + wslot) * 16;   // grid is exact; no tail

  // Wave-uniform CSR row range for rows [O, O+16): idx_out sorted, every row present.
  // readfirstlane => SGPR loop bounds: scalar loop control, no exec-masked loop exit.
  int kbase = lower_bound_i32(io, nnz, O);
  int kend  = lower_bound_i32(io, nnz, O + 16);
  kbase = __builtin_amdgcn_readfirstlane(kbase);
  const int kcnt = __builtin_amdgcn_readfirstlane(kend) - kbase;   // 16..48

  // Cooperative metadata stage: lanes 0..31 fill entry `lane`, lanes 0..19 also
  // fill entry `lane+32` (covers 52 slots).
  {
    const int l0 = lane;
    const int g0 = kbase + ((l0 < kcnt) ? l0 : (kcnt - 1));
    io_s [wslot][l0] = (l0 < kcnt) ? io[g0] : -1;
    ii_s [wslot][l0] = ii[g0];
    val_s[wslot][l0] = val[g0];
    if (lane < 20) {
      const int l1 = lane + 32;
      const int g1 = kbase + ((l1 < kcnt) ? l1 : (kcnt - 1));
      io_s [wslot][l1] = (l1 < kcnt) ? io[g1] : -1;
      ii_s [wslot][l1] = ii[g1];
      val_s[wslot][l1] = val[g1];
    }
  }
  __syncthreads();

  const int mrow = lane & 15;                 // A row (M) and B/D column (N) for this lane
  const int koff = (lane < 16) ? 0 : 2;       // K slots owned by this half-wave
  const int om   = O + mrow;                  // loop-invariant row match key

  v8f acc0 = {};   // batch cols 0..15
  v8f acc1 = {};   // batch cols 16..31

  for (int c = 0; c < kcnt; c += 4) {         // scalar (SGPR) trip count
    const int j0 = c + koff;                  // local nnz column for .x
    const int j1 = j0 + 1;                    // local nnz column for .y  (j1 <= kcnt+2 < 52)

    const int   r0 = io_s [wslot][j0];        // broadcast DS reads, no guards
    const int   r1 = io_s [wslot][j1];
    const float v0 = val_s[wslot][j0];
    const float v1 = val_s[wslot][j1];

    v2f a;
    a.x = (r0 == om) ? v0 : 0.0f;             // single v_cndmask, EXEC untouched
    a.y = (r1 == om) ? v1 : 0.0f;

    const size_t s0 = (size_t)ii_s[wslot][j0] * 32;  // gathered rows: 64B segs/half-wave
    const size_t s1 = (size_t)ii_s[wslot][j1] * 32;
    v2f b0, b1;
    b0.x = h[s0 + mrow];        b0.y = h[s1 + mrow];
    b1.x = h[s0 + 16 + mrow];   b1.y = h[s1 + 16 + mrow];

    acc0 = __builtin_amdgcn_wmma_f32_16x16x4_f32(false, a, false, b0, (short)0, acc0, false, false);
    acc1 = __builtin_amdgcn_wmma_f32_16x16x4_f32(false, a, false, b1, (short)0, acc1, false, false);
  }

  const int mbase = (lane < 16) ? 0 : 8;

  if (!tstore) {
    // Row-major store, 64B segments per half-wave; fused ReLU for hidden layers.
    // Keep RT temporal hint: next layer's gather re-reads these from L2.
#pragma unroll
    for (int r = 0; r < 8; ++r) {
      float d0 = acc0[r];
      float d1 = acc1[r];
      if (do_relu) {
        d0 = d0 > 0.0f ? d0 : 0.0f;
        d1 = d1 > 0.0f ? d1 : 0.0f;
      }
      const size_t row = (size_t)(O + mbase + r);
      out[row * 32 + mrow]      = d0;
      out[row * 32 + 16 + mrow] = d1;
    }
  } else {
    // Final layer: transpose 16x32 tile through LDS so each lane stores a
    // contiguous 64B run of d_out[b*OUTPUT + O .. O+15]. Non-temporal: d_out is
    // write-once, keep it out of L2 so layer-9 activations stay resident.
#pragma unroll
    for (int r = 0; r < 8; ++r) {
      xpose[wslot][mbase + r][mrow]      = acc0[r];
      xpose[wslot][mbase + r][16 + mrow] = acc1[r];
    }
    __syncthreads();
    float* dst = out + (size_t)lane * (size_t)out_size + O;
#pragma unroll
    for (int m4 = 0; m4 < 4; ++m4) {
      v4f q;
      q.x = xpose[wslot][m4 * 4 + 0][lane];
      q.y = xpose[wslot][m4 * 4 + 1][lane];
      q.z = xpose[wslot][m4 * 4 + 2][lane];
      q.w = xpose[wslot][m4 * 4 + 3][lane];
      __builtin_nontemporal_store(q, (v4f*)(dst + m4 * 4));
    }
  }
}

// x: [32][1024] -> xt: [1024][32]   (tiny: 128KB, one-shot)
__global__ __launch_bounds__(256)
void hnet_transpose_x(const float* __restrict__ x, float* __restrict__ xt)
{
  const int t = blockIdx.x * blockDim.x + threadIdx.x;  // 0..32767
  const int f = t >> 5;
  const int b = t & 31;
  xt[t] = x[b * 1024 + f];
}

extern "C" void kernel_launch(void* const* d_in, const int* in_sizes, int n_in,
                              void* d_out, int out_size, void* d_ws, size_t ws_size,
                              hipStream_t stream)
{
  (void)n_in; (void)out_size; (void)ws_size;
  // Input layout (setup_inputs dict order, tuples expanded):
  //   d_in[0]        : x            (32*1024 f32)
  //   d_in[1..10]    : values[i]    (f32, nnz_i)
  //   d_in[11..20]   : idx_out[i]   (i32, sorted ascending)
  //   d_in[21..30]   : idx_in[i]    (i32)
  //   d_in[31..]     : out_sizes    (unused; out_size_i == 2048<<i by construction)
  const int NL = 10;

  const float* x = (const float*)d_in[0];
  float* ws = (float*)d_ws;
  float* xt = ws;                       // 1024*32 floats (128KB), reserve 1MB
  float* W  = ws + (1u << 18);          // even-layer outputs (max 64MB -> ws needs ~65MB)
  float* D  = (float*)d_out;            // odd-layer scratch + final output (128MB)

  hnet_transpose_x<<<dim3(128), dim3(256), 0, stream>>>(x, xt);

  const float* hin = xt;
  for (int i = 0; i < NL; ++i) {
    const int    osz = 2048 << i;
    const int    nnz = in_sizes[1 + i];
    const float* v   = (const float*)d_in[1 + i];
    const int*   io  = (const int*)  d_in[11 + i];
    const int*   ii  = (const int*)  d_in[21 + i];

    float* hout;
    int relu, tstore;
    if (i == NL - 1) { hout = D;                       relu = 0; tstore = 1; }
    else             { hout = (i % 2 == 0) ? W : D;    relu = 1; tstore = 0; }

    const int tiles  = osz / 16;       // >=128, always divisible by 8
    const int blocks = tiles / 8;      // 8 waves (tiles) per 256-thread block
    hnet_spmm_wmma<<<dim3(blocks), dim3(256), 0, stream>>>(
        hin, v, io, ii, hout, osz, nnz, relu, tstore);
    hin = hout;
  }
}